// RotaryPositionalEmbedding_18622978195592
// MI455X (gfx1250) — compile-verified
//
#include <hip/hip_runtime.h>
#include <stdint.h>
#include <stddef.h>

typedef float v4f __attribute__((ext_vector_type(4)));
typedef float v2f __attribute__((ext_vector_type(2)));
typedef int   v4i_vs __attribute__((vector_size(16)));   // matches builtin param type

#define BLOCK_THREADS 256
#define WAVES_PER_BLOCK 8   // 256 / wave32
#define DK 128
#define HALF 64

// CDNA5 async global->LDS copy path (gfx1250). Guarded so both toolchains compile.
#if defined(__gfx1250__) && __has_builtin(__builtin_amdgcn_global_load_async_to_lds_b128)
#define USE_ASYNC_LDS 1
#else
#define USE_ASYNC_LDS 0
#endif

__device__ __forceinline__ void wait_asynccnt0() {
#if defined(__gfx1250__)
#if __has_builtin(__builtin_amdgcn_s_wait_asynccnt)
    __builtin_amdgcn_s_wait_asynccnt(0);
#else
    asm volatile("s_wait_asynccnt 0" ::: "memory");
#endif
#endif
}

__global__ __launch_bounds__(BLOCK_THREADS) void rope_kernel(
    const float* __restrict__ x,
    const float* __restrict__ cos_tab,
    const float* __restrict__ sin_tab,
    const int*   __restrict__ pos,
    float* __restrict__ out,
    int S, int rows)
{
    const int tid  = blockIdx.x * BLOCK_THREADS + threadIdx.x;
    const int lane = threadIdx.x & 31;      // lane within wave32
    const int w    = threadIdx.x >> 5;      // wave id within block
    const int row  = tid >> 5;              // one row (128 floats) per wave
    if (row >= rows) return;

    const int s = row % S;                  // sequence index
    const int p = pos[s];                   // gather token position (L2-resident)

#if USE_ASYNC_LDS
    // Stage this row's cos (256B) + sin (256B) into LDS with ONE async b128
    // per wave: lanes 0-15 carry cos addresses, lanes 16-31 carry sin.
    // Issued before the big x load so the DMA overlaps it (ASYNCcnt tracked).
    __shared__ float tab[WAVES_PER_BLOCK][DK];
    {
        const float* gsrc = (lane < 16)
            ? (cos_tab + (size_t)p * HALF + lane * 4)
            : (sin_tab + (size_t)p * HALF + (lane - 16) * 4);
        __builtin_amdgcn_global_load_async_to_lds_b128(
            (v4i_vs*)gsrc,
            (__attribute__((address_space(3))) v4i_vs*)&tab[w][lane * 4],
            0, 0);
    }
#endif

    // Streamed x: non-temporal b128 load (single use, keep out of caches).
    const v4f* xrow = (const v4f*)(x + (size_t)row * DK);
    const v4f xv = __builtin_nontemporal_load(xrow + lane);

    v2f c, sn;
#if USE_ASYNC_LDS
    wait_asynccnt0();
    c  = *(const v2f*)&tab[w][2 * lane];          // cos[2*lane], cos[2*lane+1]
    sn = *(const v2f*)&tab[w][HALF + 2 * lane];   // sin[2*lane], sin[2*lane+1]
#else
    c  = *(const v2f*)(cos_tab + (size_t)p * HALF + 2 * lane);
    sn = *(const v2f*)(sin_tab + (size_t)p * HALF + 2 * lane);
#endif

    v4f r;
    r.x = c.x * xv.x - sn.x * xv.y;   // pair 0: even
    r.y = sn.x * xv.x + c.x * xv.y;   // pair 0: odd
    r.z = c.y * xv.z - sn.y * xv.w;   // pair 1: even
    r.w = sn.y * xv.z + c.y * xv.w;   // pair 1: odd

    // Streamed output: non-temporal b128 store.
    __builtin_nontemporal_store(r, (v4f*)(out + (size_t)row * DK) + lane);
}

extern "C" void kernel_launch(void* const* d_in, const int* in_sizes, int n_in,
                              void* d_out, int out_size, void* d_ws, size_t ws_size,
                              hipStream_t stream) {
    const float* x       = (const float*)d_in[0];
    const float* cos_tab = (const float*)d_in[1];
    const float* sin_tab = (const float*)d_in[2];
    const int*   pos     = (const int*)d_in[3];
    float*       out     = (float*)d_out;

    const int rows = in_sizes[0] / DK;   // B*S = 131072
    const int S    = in_sizes[3];        // 16384

    const int blocks = (rows + WAVES_PER_BLOCK - 1) / WAVES_PER_BLOCK;
    rope_kernel<<<blocks, BLOCK_THREADS, 0, stream>>>(x, cos_tab, sin_tab, pos, out, S, rows);
}